// GaussianPooling_8022998908997
// MI455X (gfx1250) — compile-verified
//
#include <hip/hip_runtime.h>
#include <hip/hip_bf16.h>
#include <stdint.h>

// ---------------------------------------------------------------------------
// GaussianPooling on MI455X (gfx1250)
//
// Memory-bound gather (0.5 FLOP/byte, ~210MB traffic -> ~9us HBM roofline).
// CDNA5 path: Tensor Data Mover gathers the 5x5xC patch for one keypoint into
// LDS. Double-buffered: two TENSOR_LOAD_TO_LDS ops (256 channels each) are
// issued back-to-back; compute on chunk 0 overlaps the DMA of chunk 1 via
// s_wait_tensorcnt 1 / 0 (TDM ops complete in-order within a wave).
// ---------------------------------------------------------------------------

typedef __attribute__((ext_vector_type(4))) unsigned int u32x4;
typedef __attribute__((ext_vector_type(8))) int          i32x8;
typedef __attribute__((ext_vector_type(4))) int          i32x4;

#define HALF_K 2
#define TPB    256
#define MAXC   512
#define CHUNK  (MAXC / 2)   // 256 channels per TDM op

#if __has_builtin(__builtin_amdgcn_tensor_load_to_lds)
#define HAVE_TDM 1
#else
#define HAVE_TDM 0
#endif

#if HAVE_TDM
// Issue one TENSOR_LOAD_TO_LDS for a [tileC x 5 x 5] fp32 tile starting at
// byte address gaddr, with row stride W and channel stride H*W (elements),
// destination LDS byte offset ldsOff. Descriptor per CDNA5 ISA.
__device__ __attribute__((always_inline)) static void tdm_load_patch(
    uint64_t gaddr, uint32_t ldsOff, int tileC, int C, int H, int W, unsigned HW)
{
    // D# group 0 (128b): count=1, lds_addr, global_addr[56:0], type=2
    u32x4 g0;
    g0[0] = 1u;
    g0[1] = ldsOff;
    g0[2] = (uint32_t)gaddr;
    g0[3] = (uint32_t)((gaddr >> 32) & 0x01FFFFFFu) | (2u << 30);

    // D# group 1 (256b)
    i32x8 g1;
    g1[0] = (int)(2u << 16);                                  // wg_mask=0, data_size=4B
    g1[1] = (int)(((unsigned)W & 0xFFFFu) << 16);             // tensor_dim0 lo16 @ [63:48]
    g1[2] = (int)((((unsigned)W >> 16) & 0xFFFFu)             // tensor_dim0 hi16
                  | (((unsigned)H & 0xFFFFu) << 16));         // tensor_dim1 lo16
    g1[3] = (int)((((unsigned)H >> 16) & 0xFFFFu)             // tensor_dim1 hi16
                  | (5u << 16));                              // tile_dim0 = 5
    g1[4] = (int)(5u | ((unsigned)tileC << 16));              // tile_dim1=5, tile_dim2=tileC
    g1[5] = (int)W;                                           // tensor_dim0_stride lo32
    g1[6] = (int)((HW & 0xFFFFu) << 16);                      // stride0 hi16=0 | stride1 lo16
    g1[7] = (int)(HW >> 16);                                  // tensor_dim1_stride[47:16]

    // D# group 2 (128b): tensor_dim2=C, tensor_dim2_stride=H*W, tile_dim3=0
    i32x4 g2;
    g2[0] = C;
    g2[1] = 0;
    g2[2] = (int)HW;
    g2[3] = 0;

    // D# group 3 (128b): unused dims
    i32x4 g3 = {0, 0, 0, 0};

    // 6-arg clang-23 form: (u32x4, i32x8, i32x4, i32x4, i32x8, i32 cpol).
    // Groups 2/3 go in both the 4-dword slots and (concatenated) the 8-dword
    // slot so the descriptor is consistent under either interpretation of the
    // uncharacterized 5th operand (ignored anyway when gather_mode=0).
    i32x8 g23;
    g23[0] = g2[0]; g23[1] = g2[1]; g23[2] = g2[2]; g23[3] = g2[3];
    g23[4] = g3[0]; g23[5] = g3[1]; g23[6] = g3[2]; g23[7] = g3[3];

    __builtin_amdgcn_tensor_load_to_lds(g0, g1, g2, g3, g23, 0);
}
#endif

// Separable Gaussian stencil over 25 LDS floats per channel.
__device__ __attribute__((always_inline)) static void compute_chunk(
    const float* __restrict__ base, int nch, float* __restrict__ outp)
{
    const float g1d[5] = {0.60653065971263342f, 0.88249690258459540f, 1.0f,
                          0.88249690258459540f, 0.60653065971263342f};
    const float S     = 3.9780551245944578f;   // sum of g1d
    const float invS2 = 1.0f / (S * S);

    for (int c = threadIdx.x; c < nch; c += TPB) {
        const float* p = base + c * 25;        // stride 25 dwords: gcd(25,64)=1 -> no conflicts
        float acc = 0.0f;
        #pragma unroll
        for (int h = 0; h < 5; ++h) {
            float r = 0.0f;
            #pragma unroll
            for (int w = 0; w < 5; ++w)
                r = fmaf(g1d[w], p[h * 5 + w], r);
            acc = fmaf(g1d[h], r, acc);
        }
        outp[c] = acc * invS2;                 // coalesced b32 stores
    }
}

__global__ __launch_bounds__(TPB) void gauss_pool_tdm(
    const float* __restrict__ fm, const int* __restrict__ kp,
    float* __restrict__ out, int C, int H, int W)
{
    __shared__ float tile[2][CHUNK * 25];      // 2 x 25,600 B double buffer

    const int n = blockIdx.x;

    // keypoint coords (uniform across the block)
    int x = kp[2 * n + 0];
    int y = kp[2 * n + 1];
    x = min(max(x, HALF_K), W - HALF_K - 1);
    y = min(max(y, HALF_K), H - HALF_K - 1);

    const unsigned HW = (unsigned)(H * W);
    const int c0 = (C < CHUNK) ? C : CHUNK;    // channels in chunk 0
    const int c1 = C - c0;                     // channels in chunk 1

    const uint64_t gaddr0 = (uint64_t)(size_t)fm
                          + 4ull * ((uint64_t)(unsigned)(y - HALF_K) * (unsigned)W
                                    + (unsigned)(x - HALF_K));

#if HAVE_TDM
    if (threadIdx.x < 32) {                    // wave 0 drives the DMA
        const uint32_t ldsOff = (uint32_t)(size_t)(void*)&tile[0][0];
        tdm_load_patch(gaddr0, ldsOff, c0, C, H, W, HW);
        if (c1 > 0)
            tdm_load_patch(gaddr0 + 4ull * (uint64_t)c0 * HW,
                           ldsOff + (uint32_t)(CHUNK * 25 * 4), c1, C, H, W, HW);
#if __has_builtin(__builtin_amdgcn_s_wait_tensorcnt)
        __builtin_amdgcn_s_wait_tensorcnt(1);  // chunk 0 landed (in-order completion)
#else
        asm volatile("s_wait_tensorcnt 0x1" ::: "memory");
#endif
    }
    __syncthreads();
    compute_chunk(&tile[0][0], c0, out + (size_t)n * (unsigned)C);

    if (c1 > 0) {
        if (threadIdx.x < 32) {
#if __has_builtin(__builtin_amdgcn_s_wait_tensorcnt)
            __builtin_amdgcn_s_wait_tensorcnt(0);  // chunk 1 landed
#else
            asm volatile("s_wait_tensorcnt 0x0" ::: "memory");
#endif
        }
        __syncthreads();
        compute_chunk(&tile[1][0], c1, out + (size_t)n * (unsigned)C + c0);
    }
#else
    // ---- Fallback: direct per-thread gather, same LDS layout ----
    for (int c = threadIdx.x; c < C; c += TPB) {
        const float* src = fm + (size_t)c * HW + (size_t)(y - HALF_K) * W + (x - HALF_K);
        float* dst = &tile[c / CHUNK][(c % CHUNK) * 25];
        #pragma unroll
        for (int h = 0; h < 5; ++h)
            #pragma unroll
            for (int w = 0; w < 5; ++w)
                dst[h * 5 + w] = src[h * W + w];
    }
    __syncthreads();
    compute_chunk(&tile[0][0], c0, out + (size_t)n * (unsigned)C);
    if (c1 > 0)
        compute_chunk(&tile[1][0], c1, out + (size_t)n * (unsigned)C + c0);
#endif
}

extern "C" void kernel_launch(void* const* d_in, const int* in_sizes, int n_in,
                              void* d_out, int out_size, void* d_ws, size_t ws_size,
                              hipStream_t stream) {
    const float* fm = (const float*)d_in[0];   // [C, H, W] fp32
    const int*   kp = (const int*)d_in[1];     // [N, 2] int
    float*       out = (float*)d_out;          // [N, C] fp32

    const int H = 256, W = 256;
    const int C = in_sizes[0] / (H * W);       // 512
    const int N = in_sizes[1] / 2;             // 4096

    gauss_pool_tdm<<<dim3(N), dim3(TPB), 0, stream>>>(fm, kp, out, C, H, W);
}